// EMAAttention_8340826489510
// MI455X (gfx1250) — compile-verified
//
#include <hip/hip_runtime.h>
#include <cstdint>
#include <cmath>

// Problem constants (B,C,H,W = 16,256,80,80; K=3; ALPHA=0.1; EPS=1e-7)
#define BATCH    16
#define NCH      256
#define LROW     6400          // H*W
#define NTHREADS 256           // 8 wave32 waves per workgroup
#define PT       (LROW / NTHREADS)   // 25 elements per thread

typedef __attribute__((ext_vector_type(4))) unsigned int u32x4;
typedef __attribute__((ext_vector_type(8))) int          i32x8;
typedef __attribute__((ext_vector_type(4))) int          i32x4;

// ---------------------------------------------------------------------------
// Tensor DMA descriptor (D#) per CDNA5 ISA ch.8:
//  group0: [1:0]=count=1, [63:32]=lds_addr, [120:64]=global_addr, [127:126]=type=2
//  group1: [17:16]=data_size=2 (4B), [79:48]=tensor_dim0, [95:80]=tensor_dim1,
//          [127:112]=tile_dim0, [143:128]=tile_dim1, [207:160]=tensor_dim0_stride
// 1-D copy of LROW contiguous f32 elements.
// ---------------------------------------------------------------------------
__device__ __forceinline__ void tdm_desc_1d(u32x4& g0, i32x8& g1,
                                            uint32_t lds_off, uint64_t gaddr) {
  g0.x = 1u;                                                   // count = 1 valid descriptor
  g0.y = lds_off;                                              // LDS byte address
  g0.z = (uint32_t)gaddr;                                      // global addr [31:0]
  g0.w = ((uint32_t)(gaddr >> 32) & 0x01FFFFFFu) | (2u << 30); // addr[56:32] | type=2
  g1[0] = 0x00020000;                          // workgroup_mask=0, data_size=2 (4 bytes)
  g1[1] = (int)((LROW & 0xFFFF) << 16);        // tensor_dim0[15:0] at bits[63:48]
  g1[2] = (int)(((LROW >> 16) & 0xFFFF) | (1 << 16)); // tensor_dim0[31:16] | tensor_dim1=1
  g1[3] = (int)((uint32_t)LROW << 16);         // tile_dim0 = LROW at bits[127:112]
  g1[4] = 1;                                   // tile_dim1 = 1, tile_dim2 = 0
  g1[5] = LROW;                                // tensor_dim0_stride[31:0]
  g1[6] = 0;                                   // stride hi / dim1_stride
  g1[7] = 0;
}

__device__ __forceinline__ void tdm_load(const u32x4& g0, const i32x8& g1) {
  i32x4 z4 = {};
#if __has_include(<hip/amd_detail/amd_gfx1250_TDM.h>)
  i32x8 z8 = {};
  __builtin_amdgcn_tensor_load_to_lds(g0, g1, z4, z4, z8, 0);   // 6-arg toolchain
#else
  __builtin_amdgcn_tensor_load_to_lds(g0, g1, z4, z4, 0);       // ROCm 7.2, 5-arg
#endif
}

__device__ __forceinline__ void tdm_store(const u32x4& g0, const i32x8& g1) {
  i32x4 z4 = {};
#if __has_include(<hip/amd_detail/amd_gfx1250_TDM.h>)
  i32x8 z8 = {};
  __builtin_amdgcn_tensor_store_from_lds(g0, g1, z4, z4, z8, 0);
#else
  __builtin_amdgcn_tensor_store_from_lds(g0, g1, z4, z4, 0);
#endif
}

// ---------------------------------------------------------------------------
// Prep: wdec[l]   = ALPHA * (float)(0.9^l)            (f64 -> f32, as in reference)
//       invden[l] = 1 / ((float)(1 - 0.9^(l+1)) + EPS) (analytic cumsum of ALPHA*decay)
// ---------------------------------------------------------------------------
__global__ void ema_prep_kernel(float* __restrict__ wdec, float* __restrict__ invden) {
  int l = blockIdx.x * blockDim.x + threadIdx.x;
  if (l < LROW) {
    double d = pow(0.9, (double)l);
    wdec[l] = 0.1f * (float)d;
    double cd = 1.0 - pow(0.9, (double)(l + 1));   // sum_{i<=l} 0.1*0.9^i
    invden[l] = 1.0f / ((float)cd + 1e-7f);
  }
}

// ---------------------------------------------------------------------------
// Main: one workgroup per (b,c) row. TDM row in -> scan/conv in LDS -> TDM row out.
// ---------------------------------------------------------------------------
__global__ __launch_bounds__(NTHREADS)
void ema_attn_kernel(const float* __restrict__ x,
                     const float* __restrict__ conv_w,
                     const float* __restrict__ conv_b,
                     const float* __restrict__ wdec,
                     const float* __restrict__ invden,
                     float* __restrict__ out) {
  __shared__ float xs[LROW];                 // 25.6 KB input row
  __shared__ float os[LROW];                 // 25.6 KB output row
  __shared__ float wsum[NTHREADS / 32];      // per-wave scan partials

  const int row  = blockIdx.x;               // 0 .. B*C-1
  const int c    = row & (NCH - 1);
  const int tid  = threadIdx.x;
  const int lane = tid & 31;
  const int wid  = tid >> 5;

  const uint64_t xrow = (uint64_t)(uintptr_t)(x   + (size_t)row * LROW);
  const uint64_t orow = (uint64_t)(uintptr_t)(out + (size_t)row * LROW);

  // --- DMA the row into LDS (wave 0 only; scalar branch so EXEC-ignoring TDM
  //     is issued exactly once per workgroup) ---
  if (__builtin_amdgcn_readfirstlane(wid) == 0) {
    u32x4 g0; i32x8 g1;
    tdm_desc_1d(g0, g1, (uint32_t)(uintptr_t)&xs[0], xrow);
    tdm_load(g0, g1);
    __builtin_amdgcn_s_wait_tensorcnt(0);
  }
  __syncthreads();

  const float w0 = conv_w[3 * c + 0];
  const float w1 = conv_w[3 * c + 1];
  const float w2 = conv_w[3 * c + 2];
  const float bias = conv_b[c];

  // --- pass 1: local weighted sum over this thread's 25 contiguous elements.
  //     stride-25 LDS addressing is bank-conflict-free (gcd(25,64)==1). ---
  const int base = tid * PT;
  float wv[PT];
  float s = 0.f;
#pragma unroll
  for (int j = 0; j < PT; ++j) {
    wv[j] = wdec[base + j];                  // L2-resident, 25.6 KB table
    s = fmaf(xs[base + j], wv[j], s);
  }

  // --- wave32 inclusive scan of per-thread partials ---
  float inc = s;
#pragma unroll
  for (int d = 1; d < 32; d <<= 1) {
    float t = __shfl_up(inc, d, 32);
    if (lane >= d) inc += t;
  }
  if (lane == 31) wsum[wid] = inc;
  __syncthreads();

  float prefix = inc - s;                    // exclusive within wave
  for (int k = 0; k < wid; ++k) prefix += wsum[k];

  // --- pass 2: running EMA + depthwise conv (neighbors from LDS) + residual ---
  float run = prefix;
#pragma unroll
  for (int j = 0; j < PT; ++j) {
    const int i   = base + j;
    const float xv = xs[i];
    run = fmaf(xv, wv[j], run);              // cumsum(x * alpha * decay)
    const float ema = run * invden[i];
    const float xl = (i > 0)        ? xs[i - 1] : 0.f;
    const float xr = (i < LROW - 1) ? xs[i + 1] : 0.f;
    const float conv = fmaf(w0, xl, fmaf(w1, xv, fmaf(w2, xr, bias)));
    os[i] = ema + conv + xv;
  }
  __syncthreads();

  // --- DMA the result row back to global (wave 0 only) ---
  if (__builtin_amdgcn_readfirstlane(wid) == 0) {
    u32x4 g0; i32x8 g1;
    tdm_desc_1d(g0, g1, (uint32_t)(uintptr_t)&os[0], orow);
    tdm_store(g0, g1);
    __builtin_amdgcn_s_wait_tensorcnt(0);
  }
}

extern "C" void kernel_launch(void* const* d_in, const int* in_sizes, int n_in,
                              void* d_out, int out_size, void* d_ws, size_t ws_size,
                              hipStream_t stream) {
  (void)in_sizes; (void)n_in; (void)out_size; (void)ws_size;
  const float* x      = (const float*)d_in[0];   // [B,C,H,W] f32
  const float* conv_w = (const float*)d_in[1];   // [C,1,3]   f32
  const float* conv_b = (const float*)d_in[2];   // [C]       f32
  float* out = (float*)d_out;                    // [B,C,H,W] f32

  float* wdec   = (float*)d_ws;                  // [LROW]
  float* invden = wdec + LROW;                   // [LROW]

  ema_prep_kernel<<<(LROW + 255) / 256, 256, 0, stream>>>(wdec, invden);
  ema_attn_kernel<<<BATCH * NCH, NTHREADS, 0, stream>>>(x, conv_w, conv_b,
                                                        wdec, invden, out);
}